// SpikeDense_89034672046654
// MI455X (gfx1250) — compile-verified
//
#include <hip/hip_runtime.h>

// ---------------------------------------------------------------------------
// SpikeDense on MI455X (gfx1250): the 32 bit-plane GEMMs collapse by linearity
// to   out = relu( trunc32(x) @ W + 32*b ),   a single fp32 GEMV.
// Memory-bound: 256 MB of W @ 23.3 TB/s ~= 11 us floor. Compute (128 MFLOP)
// is free, so we run it through V_WMMA_F32_16X16X4_F32 to keep full fp32
// precision on the CDNA5 matrix pipe.
// ---------------------------------------------------------------------------

typedef __attribute__((ext_vector_type(2))) float v2f;
typedef __attribute__((ext_vector_type(8))) float v8f;

#define NN 8192                       // units (output columns)
#define KK 8192                       // in_features (reduction dim)
#define ROW_TILES 32                  // grid.y: K split for parallelism
#define ROWS_PER_TILE (KK / ROW_TILES)  // 256 rows per workgroup
#define COLS_PER_WG 128               // 8 waves * 16 columns
#define COL_TILES (NN / COLS_PER_WG)  // 64

__device__ __forceinline__ float trunc31(float v) {
    // exact fp32 truncation to 31 fractional bits == reference's greedy scan
    return floorf(v * 2147483648.0f) * (1.0f / 2147483648.0f);
}

__global__ __launch_bounds__(256) void spike_gemv_wmma(
    const float* __restrict__ x,   // [KK]
    const float* __restrict__ W,   // [KK, NN] row-major
    float* __restrict__ acc)       // [ROW_TILES, NN] partial sums
{
    const int lane = threadIdx.x & 31;
    const int wave = threadIdx.x >> 5;
    const int n16  = lane & 15;           // column within 16-wide tile
    const int half = lane >> 4;           // 0: K=0,1  1: K=2,3  (ISA 7.12.2)
    const int col0 = blockIdx.x * COLS_PER_WG + wave * 16;
    const int k0   = blockIdx.y * ROWS_PER_TILE;

    const float* Wp = W + (size_t)k0 * NN + col0 + n16;
    const float* xp = x + k0;

    v8f cacc0 = {}, cacc1 = {}, cacc2 = {}, cacc3 = {};
    const bool is_a_lane = (n16 == 0);    // only M=0 row of A carries x

    for (int kk = 0; kk < ROWS_PER_TILE; kk += 16) {
        // 4 independent accumulator chains per 16 K-rows
        {
            const int kr = kk + 0 + 2 * half;
            float a0 = trunc31(xp[kr]);
            float a1 = trunc31(xp[kr + 1]);
            a0 = is_a_lane ? a0 : 0.0f;
            a1 = is_a_lane ? a1 : 0.0f;
            v2f A; A.x = a0; A.y = a1;
            v2f B; B.x = Wp[(size_t)kr * NN]; B.y = Wp[(size_t)(kr + 1) * NN];
            cacc0 = __builtin_amdgcn_wmma_f32_16x16x4_f32(
                false, A, false, B, (short)0, cacc0, false, false);
        }
        {
            const int kr = kk + 4 + 2 * half;
            float a0 = trunc31(xp[kr]);
            float a1 = trunc31(xp[kr + 1]);
            a0 = is_a_lane ? a0 : 0.0f;
            a1 = is_a_lane ? a1 : 0.0f;
            v2f A; A.x = a0; A.y = a1;
            v2f B; B.x = Wp[(size_t)kr * NN]; B.y = Wp[(size_t)(kr + 1) * NN];
            cacc1 = __builtin_amdgcn_wmma_f32_16x16x4_f32(
                false, A, false, B, (short)0, cacc1, false, false);
        }
        {
            const int kr = kk + 8 + 2 * half;
            float a0 = trunc31(xp[kr]);
            float a1 = trunc31(xp[kr + 1]);
            a0 = is_a_lane ? a0 : 0.0f;
            a1 = is_a_lane ? a1 : 0.0f;
            v2f A; A.x = a0; A.y = a1;
            v2f B; B.x = Wp[(size_t)kr * NN]; B.y = Wp[(size_t)(kr + 1) * NN];
            cacc2 = __builtin_amdgcn_wmma_f32_16x16x4_f32(
                false, A, false, B, (short)0, cacc2, false, false);
        }
        {
            const int kr = kk + 12 + 2 * half;
            float a0 = trunc31(xp[kr]);
            float a1 = trunc31(xp[kr + 1]);
            a0 = is_a_lane ? a0 : 0.0f;
            a1 = is_a_lane ? a1 : 0.0f;
            v2f A; A.x = a0; A.y = a1;
            v2f B; B.x = Wp[(size_t)kr * NN]; B.y = Wp[(size_t)(kr + 1) * NN];
            cacc3 = __builtin_amdgcn_wmma_f32_16x16x4_f32(
                false, A, false, B, (short)0, cacc3, false, false);
        }
    }

    v8f c = (cacc0 + cacc1) + (cacc2 + cacc3);

    // D row M=0 lives in VGPR0 of lanes 0..15 (N = lane). One writer per slot:
    // fully deterministic, no atomics.
    if (lane < 16) {
        acc[(size_t)blockIdx.y * NN + col0 + lane] = c[0];
    }
}

__global__ __launch_bounds__(256) void spike_finalize(
    const float* __restrict__ acc,   // [ROW_TILES, NN]
    const float* __restrict__ b,     // [NN]
    float* __restrict__ out)         // [NN]
{
    const int j = blockIdx.x * blockDim.x + threadIdx.x;
    if (j < NN) {
        float s = 32.0f * b[j];      // bias added once per bit-plane in ref
        #pragma unroll
        for (int r = 0; r < ROW_TILES; ++r) {
            s += acc[(size_t)r * NN + j];
        }
        out[j] = s > 0.0f ? s : 0.0f;
    }
}

extern "C" void kernel_launch(void* const* d_in, const int* in_sizes, int n_in,
                              void* d_out, int out_size, void* d_ws, size_t ws_size,
                              hipStream_t stream) {
    const float* x = (const float*)d_in[0];   // [1, 8192]
    const float* W = (const float*)d_in[1];   // [8192, 8192]
    const float* b = (const float*)d_in[2];   // [8192]
    float* out = (float*)d_out;               // [1, 8192]
    float* acc = (float*)d_ws;                // ROW_TILES*NN floats = 1 MB

    dim3 grid(COL_TILES, ROW_TILES);
    spike_gemv_wmma<<<grid, 256, 0, stream>>>(x, W, acc);
    spike_finalize<<<(NN + 255) / 256, 256, 0, stream>>>(acc, b, out);
}